// ResampleFromUV_58463094833219
// MI455X (gfx1250) — compile-verified
//
#include <hip/hip_runtime.h>

// Problem constants from the reference (fixed shapes).
constexpr int B  = 2;
constexpr int C  = 16;
constexpr int N  = 512;
constexpr int Ht = 64;
constexpr int Wt = 64;
constexpr int Ho = 1024;
constexpr int Wo = 1024;

constexpr int      BC    = B * C;          // 32 channel planes
constexpr unsigned SLICE = N * Ht * Wt;    // 2,097,152 floats per (b,c) atlas slice (8 MiB)
constexpr unsigned OUTP  = Ho * Wo;        // 1,048,576 output pixels per plane

typedef float v2f __attribute__((ext_vector_type(2)));
// 4-byte-aligned pair: gfx1250 global memory allows dword-aligned b64 loads,
// so this should still lower to a single global_load_b64 per row pair.
typedef v2f v2f_a4 __attribute__((aligned(4)));

static __device__ __forceinline__ unsigned umin_u32(unsigned a, unsigned b) {
    return a < b ? a : b;   // -> v_min_u32
}
static __device__ __forceinline__ int max_i32(int a, int b) {
    return a > b ? a : b;   // -> v_max_i32
}

__global__ __launch_bounds__(256) void ResampleFromUV_kernel(
    const float* __restrict__ x,        // [B,C,N,Ht,Wt]
    const int*   __restrict__ quad_idx, // [Ho,Wo] (int32 after jax canonicalization)
    const float* __restrict__ tex_uv,   // [Ho,Wo,2]
    float*       __restrict__ out)      // [B,C,Ho,Wo]
{
    const unsigned pix = blockIdx.x * 256u + threadIdx.x;
    if (pix >= OUTP) return;

    // ---- per-pixel data: streamed once -> non-temporal loads (don't pollute L2) ----
    const unsigned n = (unsigned)__builtin_nontemporal_load(quad_idx + pix);
    const v2f uv = __builtin_nontemporal_load(reinterpret_cast<const v2f*>(tex_uv) + pix);

    // ---- bilinear coordinates (computed once, reused across all 32 channel planes) ----
    const float u = uv.x * (float)(Wt - 1);
    const float v = uv.y * (float)(Ht - 1);
    const float x0f = floorf(u);
    const float y0f = floorf(v);
    const float wu = u - x0f;
    const float wv = v - y0f;

    // Clamp exactly like the reference.
    const unsigned x0 = umin_u32((unsigned)max_i32((int)x0f, 0), (unsigned)(Wt - 1));
    const unsigned y0 = umin_u32((unsigned)max_i32((int)y0f, 0), (unsigned)(Ht - 1));
    const unsigned y1 = umin_u32(y0 + 1u, (unsigned)(Ht - 1));

    // Horizontal tap pair starts at xp0 so (xp0, xp0+1) is one b64 load.
    // Normal case (x0 <= 62): pair = (x0, x1), weights (1-wu, wu).
    // Edge case  (x0 == 63): pair = (62, 63), both reference taps are elem 63
    //                        -> weights (0, 1). In both cases wa + wb == 1.
    const unsigned xp0 = umin_u32(x0, (unsigned)(Wt - 2));
    const float ru = 1.0f - wu;
    const float wa = (x0 == (unsigned)(Wt - 1)) ? 0.0f : ru;
    const float wb = 1.0f - wa;

    // pair offsets inside one (b,c) slice (identical for every slice)
    const unsigned base = n << 12;            // n * Ht*Wt
    const unsigned o0   = base + (y0 << 6) + xp0;   // top row pair
    const unsigned o1   = base + (y1 << 6) + xp0;   // bottom row pair

    const float rv  = 1.0f - wv;
    const float w00 = wa * rv;
    const float w01 = wb * rv;
    const float w10 = wa * wv;
    const float w11 = wb * wv;

    // ---- sweep the 32 channel planes: 2 x b64 pair-gathers per plane (one
    //      cacheline touch per row ~97% of the time), all hitting the 192 MB L2
    //      (8 MiB slice, ~64 reuses/line). Stores are coalesced + non-temporal
    //      so the 128 MiB output stream never evicts the atlas from L2. ----
    const float* __restrict__ xp = x;
    float*       __restrict__ op = out + pix;

#pragma unroll 4
    for (int bc = 0; bc < BC; ++bc) {
        const v2f t = *reinterpret_cast<const v2f_a4*>(xp + o0);
        const v2f b = *reinterpret_cast<const v2f_a4*>(xp + o1);
        const float r = t.x * w00 + t.y * w01 + b.x * w10 + b.y * w11;
        __builtin_nontemporal_store(r, op);
        xp += SLICE;
        op += OUTP;
    }
}

extern "C" void kernel_launch(void* const* d_in, const int* in_sizes, int n_in,
                              void* d_out, int out_size, void* d_ws, size_t ws_size,
                              hipStream_t stream) {
    const float* x        = (const float*)d_in[0];
    const int*   quad_idx = (const int*)d_in[1];
    const float* tex_uv   = (const float*)d_in[2];
    float*       out      = (float*)d_out;

    const int threads = 256;                       // 8 wave32 waves per block
    const int blocks  = (int)(OUTP / threads);     // 4096 blocks, exact cover
    ResampleFromUV_kernel<<<blocks, threads, 0, stream>>>(x, quad_idx, tex_uv, out);
}